// Model_47107201302874
// MI455X (gfx1250) — compile-verified
//
#include <hip/hip_runtime.h>
#include <math.h>

typedef __attribute__((ext_vector_type(2))) float v2f;
typedef __attribute__((ext_vector_type(8))) float v8f;

#define NB 1024   // blocks; each covers 2^16 contiguous elements (bits 16..25 = blockIdx)
#define NT 256    // threads; 8 wave32s

// Pass 1: per-block partial sums of (phi - lam)^2 using Kronecker factorization.
__global__ __launch_bounds__(NT) void phi_loss_partial(
    const float* __restrict__ theta,
    const float* __restrict__ lam,
    float* __restrict__ partial)
{
  __shared__ float fb[26][2];   // fb[b][t]: bit b of index, t=0 -> sin|theta[25-b]|, t=1 -> cos
  __shared__ float lutIter[64]; // product over bits 10..15
  __shared__ float accLds[NT];
  __shared__ float red[32];

  const int tid = threadIdx.x;
  const int bid = blockIdx.x;

  if (tid < 26) {
    float a = fabsf(theta[tid]);
    int b = 25 - tid;           // theta[0] maps to MSB (bit 25)
    fb[b][0] = sinf(a);
    fb[b][1] = cosf(a);
  }
  __syncthreads();

  if (tid < 64) {
    float p = 1.0f;
#pragma unroll
    for (int b = 10; b <= 15; ++b) p *= fb[b][(tid >> (b - 10)) & 1];
    lutIter[tid] = p;
  }

  // Per-thread constant: bits 2..9 come from tid, bits 16..25 from blockIdx.
  float baseT = 1.0f;
#pragma unroll
  for (int b = 2; b <= 9; ++b)  baseT *= fb[b][(tid >> (b - 2)) & 1];
#pragma unroll
  for (int b = 16; b <= 25; ++b) baseT *= fb[b][(bid >> (b - 16)) & 1];

  // bits 0..1 live inside each float4: fold into 4 per-thread constants
  float g0 = baseT * fb[1][0] * fb[0][0];
  float g1 = baseT * fb[1][0] * fb[0][1];
  float g2 = baseT * fb[1][1] * fb[0][0];
  float g3 = baseT * fb[1][1] * fb[0][1];

  __syncthreads();

  // element e = (bid<<16) + iter*1024 + tid*4  -> float4 index (bid<<14) + iter*256 + tid
  const float4* lam4 = (const float4*)lam + ((size_t)bid << 14) + tid;
  float acc = 0.0f;
#pragma unroll 4
  for (int iter = 0; iter < 64; ++iter) {
    __builtin_prefetch(lam4 + iter * 256 + 2048, 0, 1);  // global_prefetch_b8, 8 iters ahead
    float4 l = lam4[iter * 256];                         // coalesced global_load_b128
    float m  = lutIter[iter];                            // bits 10..15 product
    float t0 = fmaf(m, g0, -l.x);
    float t1 = fmaf(m, g1, -l.y);
    float t2 = fmaf(m, g2, -l.z);
    float t3 = fmaf(m, g3, -l.w);
    acc = fmaf(t0, t0, acc);
    acc = fmaf(t1, t1, acc);
    acc = fmaf(t2, t2, acc);
    acc = fmaf(t3, t3, acc);
  }

  accLds[tid] = acc;
  __syncthreads();

  // Wave 0: 32-lane reduction via V_WMMA_F32_16X16X4_F32 (exact f32, deterministic).
  // A(16x4): lane n<16 -> A[n][0]=v, lane 16+n -> A[n][2]=v, K=1,3 zero.
  // B(4x16) = ones. D[m][n] = v_m + v_{m+16} replicated over n.
  if (tid < 32) {
    float v = 0.0f;
#pragma unroll
    for (int w = 0; w < 8; ++w) v += accLds[tid + 32 * w];
    v2f a  = {v, 0.0f};
    v2f bb = {1.0f, 1.0f};
    v8f c  = {};
    v8f dm = __builtin_amdgcn_wmma_f32_16x16x4_f32(
        /*neg_a=*/false, a, /*neg_b=*/false, bb,
        /*c_mod=*/(short)0, c, /*reuse_a=*/false, /*reuse_b=*/false);
    float loc = dm[0] + dm[1] + dm[2] + dm[3] + dm[4] + dm[5] + dm[6] + dm[7];
    red[tid] = loc;
  }
  __syncthreads();
  if (tid == 0) partial[bid] = red[0] + red[16];  // sums all 32 lane values
}

// Pass 2: deterministic tree reduction of the 1024 block partials.
__global__ __launch_bounds__(256) void reduce_partials(
    const float* __restrict__ ws, float* __restrict__ out)
{
  __shared__ float s[256];
  const int t = threadIdx.x;
  float v = ws[t] + ws[t + 256] + ws[t + 512] + ws[t + 768];
  s[t] = v;
  __syncthreads();
#pragma unroll
  for (int k = 128; k > 0; k >>= 1) {
    if (t < k) s[t] += s[t + k];
    __syncthreads();
  }
  if (t == 0) out[0] = s[0];
}

extern "C" void kernel_launch(void* const* d_in, const int* in_sizes, int n_in,
                              void* d_out, int out_size, void* d_ws, size_t ws_size,
                              hipStream_t stream) {
  const float* theta = (const float*)d_in[0];  // [26]
  const float* lam   = (const float*)d_in[1];  // [2^26]
  float* out = (float*)d_out;                  // [1], f32
  float* ws  = (float*)d_ws;                   // >= 1024 floats of scratch

  phi_loss_partial<<<NB, NT, 0, stream>>>(theta, lam, ws);
  reduce_partials<<<1, 256, 0, stream>>>(ws, out);
}